// PCT_49469433315920
// MI455X (gfx1250) — compile-verified
//
#include <hip/hip_runtime.h>
#include <hip/hip_bf16.h>
#include <math.h>

typedef float v2f __attribute__((ext_vector_type(2)));
typedef float v8f __attribute__((ext_vector_type(8)));

#define NB 4096
#define NC 128
#define NCP 129              // LDS row pitch: conflict-free column walks
#define BC (NB * NC)
#define EPS_BN 1e-5f
#define EPS_ATT 1e-9f

__device__ __forceinline__ float sigmoidf_(float x) {
    return 1.0f / (1.0f + __expf(-x));
}

// ---------------------------------------------------------------------------
// K0: zero per-channel BN accumulators (must be re-zeroed every launch)
// ---------------------------------------------------------------------------
__global__ void pct_init_kernel(float* __restrict__ cs, float* __restrict__ css) {
    int i = threadIdx.x;
    if (i < NC) { cs[i] = 0.0f; css[i] = 0.0f; }
}

// ---------------------------------------------------------------------------
// K1: per-batch attention block. One 128-thread block per batch element.
// Materializes E[c,d] = exp(x1[c]*x2[d] - m_c) once in LDS (66KB, pitch 129
// => column reads hit distinct banks), so every element needs exactly ONE
// v_exp_f32 instead of four. All reductions become LDS row/column sweeps.
// Outputs: x_r[b,c] and gate[b,c]*x1[b,c].
// ---------------------------------------------------------------------------
__global__ void pct_attn_kernel(const float* __restrict__ x1,
                                const float* __restrict__ x2,
                                const float* __restrict__ x1_2,
                                const float* __restrict__ fcw,   // 257 floats
                                const float* __restrict__ fcb,   // 1 float
                                float* __restrict__ xr_out,      // [NB*NC]
                                float* __restrict__ gx1_out) {   // [NB*NC]
    __shared__ float sE[NC][NCP];                       // 66048 B
    __shared__ float sx1[NC], sx2[NC], sx12[NC];
    __shared__ float sInvZ[NC], sU[NC], sW1[NC], sW2Z[NC];

    const int b = blockIdx.x;
    const int c = threadIdx.x;          // 0..127
    const int base = b * NC;

    sx1[c]  = x1[base + c];
    sx2[c]  = x2[base + c];
    sx12[c] = x1_2[base + c];
    sW1[c]  = fcw[c];
    __syncthreads();

    const float x1c = sx1[c];

    // row max of x1[c]*x2[d] over d (softmax stabilizer), via min/max of x2
    float mx = -INFINITY, mn = INFINITY;
    for (int d = 0; d < NC; ++d) {
        float v = sx2[d];
        mx = fmaxf(mx, v);
        mn = fminf(mn, v);
    }
    const float m_c = fmaxf(x1c * mx, x1c * mn);

    // Pass 1 (row): fill E row c, accumulate Z_c.  One exp per element.
    float z = 0.0f;
    for (int d = 0; d < NC; ++d) {
        float e = __expf(x1c * sx2[d] - m_c);
        sE[c][d] = e;
        z += e;
    }
    const float invz = 1.0f / z;
    sInvZ[c] = invz;
    __syncthreads();

    // Pass 2 (column c): S_c = sum_d energy[d,c] = sum_d E[d,c]/Z_d
    float S = 0.0f;
    for (int d = 0; d < NC; ++d)
        S += sE[d][c] * sInvZ[d];
    sU[c]   = sx12[c] / (EPS_ATT + S);      // x1_2 / (eps + colsum)
    sW2Z[c] = fcw[NC + c] * invz;           // w2[c] / Z_c
    __syncthreads();

    // Pass 3 (row): x_r[c] = invz * sum_d E*u[d];  g1 = invz * sum_d E*w1[d]
    float xr = 0.0f, g1 = 0.0f;
    for (int d = 0; d < NC; ++d) {
        float e = sE[c][d];
        xr += e * sU[d];
        g1 += e * sW1[d];
    }
    xr *= invz;
    g1 *= invz;

    // Pass 4 (column c): g2[c] = sum_d (E[d,c]/Z_d) * w2[d]
    float g2 = 0.0f;
    for (int d = 0; d < NC; ++d)
        g2 += sE[d][c] * sW2Z[d];

    const float gate = sigmoidf_(g1 + g2 + fcw[2 * NC] * x1c + fcb[0]);
    xr_out[base + c]  = xr;
    gx1_out[base + c] = gate * x1c;
}

// ---------------------------------------------------------------------------
// K2: GEMM  t = X_R(4096x128) @ conv_w^T(128x128)  via V_WMMA_F32_16X16X4_F32.
// Block = 256 threads (8 waves); block handles 16 batches x 128 channels;
// wave w owns the 16x16 tile at channel offset 16*w. Fuses y = x1 - (t+bias),
// stores Y, and accumulates per-channel sum / sumsq for BatchNorm.
// ---------------------------------------------------------------------------
__global__ void pct_gemm_kernel(const float* __restrict__ xr,     // [NB*NC]
                                const float* __restrict__ W,      // conv_w [NC*NC] row-major (o,c)
                                const float* __restrict__ convb,  // [NC]
                                const float* __restrict__ x1,
                                float* __restrict__ Y,            // [NB*NC]
                                float* __restrict__ cs,
                                float* __restrict__ css) {
    const int wave  = threadIdx.x >> 5;   // 0..7 -> channel tile
    const int lane  = threadIdx.x & 31;
    const int b0    = blockIdx.x * 16;    // batch tile origin
    const int n0    = wave * 16;          // channel tile origin
    const int l16   = lane & 15;
    const int khalf = (lane >> 4) * 2;    // lanes 0-15 -> K{0,1}; lanes 16-31 -> K{2,3}

    // A: 16x4 f32 tile of X_R rows (batches); B: 4x16 tile of W^T, i.e. B[k][n] = W[n][k]
    const float* arow = xr + (size_t)(b0 + l16) * NC + khalf;
    const float* brow = W  + (size_t)(n0 + l16) * NC + khalf;

    v8f acc = {};
    #pragma unroll 4
    for (int k0 = 0; k0 < NC; k0 += 4) {
        v2f a, bm;
        a.x  = arow[k0];
        a.y  = arow[k0 + 1];
        bm.x = brow[k0];
        bm.y = brow[k0 + 1];
        acc = __builtin_amdgcn_wmma_f32_16x16x4_f32(false, a, false, bm,
                                                    (short)0, acc, false, false);
    }

    // D layout: VGPR r -> rows (m=r | m=8+r); lane<16 covers m=r, lane>=16 covers m=8+r
    const int mbase = (lane >> 4) * 8;
    const int ch    = n0 + l16;
    const float cb  = convb[ch];
    float s = 0.0f, s2 = 0.0f;
    #pragma unroll
    for (int r = 0; r < 8; ++r) {
        int bb  = b0 + mbase + r;
        float y = x1[(size_t)bb * NC + ch] - (acc[r] + cb);
        Y[(size_t)bb * NC + ch] = y;
        s  += y;
        s2 += y * y;
    }
    atomicAdd(&cs[ch],  s);
    atomicAdd(&css[ch], s2);
}

// ---------------------------------------------------------------------------
// K3: finalize  out = gate*x1 + relu( BN(y) )
// ---------------------------------------------------------------------------
__global__ void pct_final_kernel(const float* __restrict__ Y,
                                 const float* __restrict__ gx1,
                                 const float* __restrict__ cs,
                                 const float* __restrict__ css,
                                 const float* __restrict__ gamma,
                                 const float* __restrict__ beta,
                                 float* __restrict__ out) {
    const int i = blockIdx.x * blockDim.x + threadIdx.x;
    if (i >= BC) return;
    const int c = i & (NC - 1);
    const float invN = 1.0f / (float)NB;
    const float mean = cs[c] * invN;
    const float var  = css[c] * invN - mean * mean;   // biased, matches jnp.var
    const float yn   = (Y[i] - mean) * __frsqrt_rn(var + EPS_BN) * gamma[c] + beta[c];
    out[i] = gx1[i] + fmaxf(yn, 0.0f);
}

// ---------------------------------------------------------------------------
extern "C" void kernel_launch(void* const* d_in, const int* in_sizes, int n_in,
                              void* d_out, int out_size, void* d_ws, size_t ws_size,
                              hipStream_t stream) {
    (void)in_sizes; (void)n_in; (void)out_size; (void)ws_size;
    const float* x1     = (const float*)d_in[0];
    const float* x2     = (const float*)d_in[1];
    const float* x1_2   = (const float*)d_in[2];
    const float* conv_w = (const float*)d_in[3];
    const float* conv_b = (const float*)d_in[4];
    const float* gamma  = (const float*)d_in[5];
    const float* beta   = (const float*)d_in[6];
    const float* fc_w   = (const float*)d_in[7];
    const float* fc_b   = (const float*)d_in[8];
    float* out = (float*)d_out;

    float* ws  = (float*)d_ws;
    float* xr  = ws;               // [BC]
    float* gx1 = ws + BC;          // [BC]
    float* Y   = ws + 2 * BC;      // [BC]
    float* cs  = ws + 3 * BC;      // [NC]
    float* css = cs + NC;          // [NC]

    pct_init_kernel<<<1, 256, 0, stream>>>(cs, css);
    pct_attn_kernel<<<NB, NC, 0, stream>>>(x1, x2, x1_2, fc_w, fc_b, xr, gx1);
    pct_gemm_kernel<<<NB / 16, 256, 0, stream>>>(xr, conv_w, conv_b, x1, Y, cs, css);
    pct_final_kernel<<<BC / 256, 256, 0, stream>>>(Y, gx1, cs, css, gamma, beta, out);
}